// ModularGNN_25572235281175
// MI455X (gfx1250) — compile-verified
//
#include <hip/hip_runtime.h>

#define DD 128
#define EPSV 1e-5f

typedef __attribute__((ext_vector_type(2))) float v2f;
typedef __attribute__((ext_vector_type(8))) float v8f;
typedef __attribute__((ext_vector_type(4))) unsigned int u32x4;
typedef __attribute__((ext_vector_type(8))) int i32x8;
typedef __attribute__((ext_vector_type(4))) int i32x4;

// ---------------- GEMM: out[n,128] = A[n,128] @ W[128,128] ----------------
// Optional fused epilogues:
//   bias/do_relu        : out = relu(out + bias)
//   dinv/selfbias/agg   : also write agg = out * dinv[row]^2 + selfbias  (GCN self-loop init)
__global__ __launch_bounds__(256)
void gemm_wmma_f32(const float* __restrict__ A, const float* __restrict__ W,
                   const float* __restrict__ bias, float* __restrict__ out,
                   int nrows, int do_relu,
                   const float* __restrict__ dinv, const float* __restrict__ selfbias,
                   float* __restrict__ agg)
{
    __shared__ float As[16 * DD];
    const int block_row = blockIdx.x * 16;
    if (block_row >= nrows) return;
    const int tid = threadIdx.x;

    // Stage the contiguous 16x128 fp32 A-panel (8 KB) into LDS with the TDM.
    // One wave issues the DMA; OOB elements (tail tiles) read back as zero.
    if (tid < 32) {
        unsigned long long gaddr = (unsigned long long)(const void*)(A + (size_t)block_row * DD);
        unsigned int lds_off = (unsigned int)(size_t)(void*)As;
        unsigned int avail = (unsigned int)(nrows - block_row) * DD;
        if (avail > 16u * DD) avail = 16u * DD;
        const unsigned int tile0 = 16u * DD;     // 2048 elements

        u32x4 g0;
        g0[0] = 1u;                                           // count=1 (valid user D#)
        g0[1] = lds_off;                                      // lds_addr (bytes)
        g0[2] = (unsigned int)gaddr;                          // global_addr[31:0]
        g0[3] = ((unsigned int)(gaddr >> 32) & 0x1FFFFFFu)    // global_addr[56:32]
              | (2u << 30);                                   // type=2 ("image")
        i32x8 g1;
        g1[0] = (int)(2u << 16);                              // data_size=2 -> 4 bytes
        g1[1] = (int)((avail & 0xFFFFu) << 16);               // tensor_dim0[15:0]
        g1[2] = (int)(((avail >> 16) & 0xFFFFu)               // tensor_dim0[31:16]
              | (1u << 16));                                  // tensor_dim1 = 1
        g1[3] = (int)((tile0 & 0xFFFFu) << 16);               // tile_dim0 = 2048
        g1[4] = (int)1;                                       // tile_dim1 = 1
        g1[5] = (int)tile0;                                   // tensor_dim0_stride = 2048
        g1[6] = 0;
        g1[7] = 0;
        i32x4 gz4 = {0, 0, 0, 0};                             // groups 2/3 unused (<=2D)
        i32x8 gz8 = {0, 0, 0, 0, 0, 0, 0, 0};                 // extra group (clang-23 form)
        __builtin_amdgcn_tensor_load_to_lds(g0, g1, gz4, gz4, gz8, 0);
        __builtin_amdgcn_s_wait_tensorcnt(0);
    }
    __syncthreads();

    const int wave = tid >> 5;      // 8 waves -> 8 column tiles of 16
    const int lane = tid & 31;
    const int m    = lane & 15;
    const int half = lane >> 4;
    const int col0 = wave * 16;
    v8f c = {};
    for (int k = 0; k < DD; k += 4) {
        v2f a, b;
        a[0] = As[m * DD + k + 2 * half];
        a[1] = As[m * DD + k + 2 * half + 1];
        b[0] = W[(size_t)(k + 2 * half) * DD + col0 + m];
        b[1] = W[(size_t)(k + 2 * half + 1) * DD + col0 + m];
        c = __builtin_amdgcn_wmma_f32_16x16x4_f32(false, a, false, b, (short)0, c, false, false);
    }
    const int col = col0 + m;
    const float badd = bias ? bias[col] : 0.0f;
    const float sbadd = selfbias ? selfbias[col] : 0.0f;
    for (int r = 0; r < 8; ++r) {
        int row = block_row + r + 8 * half;
        if (row >= nrows) break;
        float v = c[r] + badd;
        if (do_relu) v = fmaxf(v, 0.0f);
        out[(size_t)row * DD + col] = v;
        if (agg) {                                  // fused GCN self-loop + bias
            float di = dinv[row];
            agg[(size_t)row * DD + col] = v * di * di + sbadd;
        }
    }
}

// ---------------- degree / norm ----------------
__global__ void k_init_deg(float* deg, int n) {
    int i = blockIdx.x * 256 + threadIdx.x;
    if (i < n) deg[i] = 1.0f;                      // self-loop contribution
}
__global__ void k_add_deg(const int* __restrict__ col, float* deg, int e) {
    int i = blockIdx.x * 256 + threadIdx.x;
    if (i < e) atomicAdd(&deg[col[i]], 1.0f);
}
__global__ void k_make_dinv(float* deg, int n) {
    int i = blockIdx.x * 256 + threadIdx.x;
    if (i < n) { float d = deg[i]; deg[i] = (d > 0.0f) ? rsqrtf(d) : 0.0f; }
}
__global__ void k_make_norm(const int* __restrict__ row, const int* __restrict__ col,
                            const float* __restrict__ dinv, float* __restrict__ norm, int e) {
    int i = blockIdx.x * 256 + threadIdx.x;
    if (i < e) norm[i] = dinv[row[i]] * dinv[col[i]];
}

// ---------------- edge scatter: one wave per edge, 4 float atomics per lane ----------------
__global__ __launch_bounds__(256)
void k_edge_scatter(const int* __restrict__ row, const int* __restrict__ col,
                    const float* __restrict__ norm, const float* __restrict__ hw,
                    float* __restrict__ agg, int e) {
    int edge = blockIdx.x * 8 + (threadIdx.x >> 5);
    if (edge >= e) return;
    int lane = threadIdx.x & 31;
    int r = row[edge], c = col[edge];
    float w = norm[edge];
    float4 v = *(const float4*)(hw + (size_t)r * DD + lane * 4);
    float* dst = agg + (size_t)c * DD + lane * 4;
    atomicAdd(dst + 0, v.x * w);
    atomicAdd(dst + 1, v.y * w);
    atomicAdd(dst + 2, v.z * w);
    atomicAdd(dst + 3, v.w * w);
}

// ---------------- graph norm ----------------
__global__ void k_zero(float* v, int n) {
    int i = blockIdx.x * 256 + threadIdx.x;
    if (i < n) v[i] = 0.0f;
}
__global__ __launch_bounds__(256)
void k_colsum(const float* __restrict__ h, float* __restrict__ colsum, int n) {
    int c  = threadIdx.x & 127;
    int r0 = blockIdx.x * 64 + (threadIdx.x >> 7);
    int rmax = blockIdx.x * 64 + 64; if (rmax > n) rmax = n;
    float s = 0.0f;
    for (int r = r0; r < rmax; r += 2) s += h[(size_t)r * DD + c];
    __shared__ float sh[256];
    sh[threadIdx.x] = s;
    __syncthreads();
    if (threadIdx.x < 128) atomicAdd(&colsum[c], sh[threadIdx.x] + sh[threadIdx.x + 128]);
}
// h -= mean*ms (in place); accumulate sum of squares per column
__global__ __launch_bounds__(256)
void k_center(float* __restrict__ h, const float* __restrict__ colsum,
              const float* __restrict__ ms, float* __restrict__ colsum2,
              int n, float inv_n) {
    int c  = threadIdx.x & 127;
    int r0 = blockIdx.x * 64 + (threadIdx.x >> 7);
    int rmax = blockIdx.x * 64 + 64; if (rmax > n) rmax = n;
    float sub = colsum[c] * inv_n * ms[c];
    float s2 = 0.0f;
    for (int r = r0; r < rmax; r += 2) {
        size_t idx = (size_t)r * DD + c;
        float v = h[idx] - sub;
        h[idx] = v;
        s2 += v * v;
    }
    __shared__ float sh[256];
    sh[threadIdx.x] = s2;
    __syncthreads();
    if (threadIdx.x < 128) atomicAdd(&colsum2[c], sh[threadIdx.x] + sh[threadIdx.x + 128]);
}
// x_out = relu(h * rsqrt(var+eps) * w + b) + residual   (in-place safe per element)
__global__ void k_finalize(const float* __restrict__ h, const float* __restrict__ colsum2,
                           const float* __restrict__ w, const float* __restrict__ b,
                           const float* __restrict__ residual, float* __restrict__ xout,
                           int total, float inv_n) {
    int idx = blockIdx.x * 256 + threadIdx.x;
    if (idx >= total) return;
    int c = idx & 127;
    float var = colsum2[c] * inv_n;
    float v = h[idx] * rsqrtf(var + EPSV) * w[c] + b[c];
    v = fmaxf(v, 0.0f);
    xout[idx] = v + residual[idx];
}

// ---------------- head: out[n] = dot(x[n,:], hw) + hb ----------------
__global__ __launch_bounds__(256)
void k_head(const float* __restrict__ X, const float* __restrict__ hw,
            const float* __restrict__ hb, float* __restrict__ out, int n) {
    int row  = blockIdx.x * 8 + (threadIdx.x >> 5);
    int lane = threadIdx.x & 31;
    if (row >= n) return;
    const float* x = X + (size_t)row * DD;
    float s = 0.0f;
    for (int i = lane; i < DD; i += 32) s += x[i] * hw[i];
    for (int off = 16; off; off >>= 1) s += __shfl_xor(s, off, 32);
    if (lane == 0) out[row] = s + hb[0];
}

extern "C" void kernel_launch(void* const* d_in, const int* in_sizes, int n_in,
                              void* d_out, int out_size, void* d_ws, size_t ws_size,
                              hipStream_t stream) {
    const float* xin    = (const float*)d_in[0];
    const int*   eidx   = (const int*)d_in[1];
    const float* conv_w = (const float*)d_in[2];
    const float* conv_b = (const float*)d_in[3];
    const float* gn_w   = (const float*)d_in[4];
    const float* gn_b   = (const float*)d_in[5];
    const float* gn_ms  = (const float*)d_in[6];
    const float* proj_w = (const float*)d_in[7];
    const float* mlp_w  = (const float*)d_in[8];
    const float* mlp_b  = (const float*)d_in[9];
    const float* head_w = (const float*)d_in[10];
    const float* head_b = (const float*)d_in[11];
    float* out = (float*)d_out;

    const int N = in_sizes[0] / DD;
    const int E = in_sizes[1] / 2;
    const int L = in_sizes[2] / (DD * DD);
    const int total = N * DD;
    const float inv_n = 1.0f / (float)N;

    const int* erow = eidx;
    const int* ecol = eidx + E;

    // workspace layout (floats)
    float* ws   = (float*)d_ws;
    float* hw   = ws;                  // N*D
    float* agg  = hw  + (size_t)total; // N*D
    float* xcur = agg + (size_t)total; // N*D
    float* tmp  = xcur + (size_t)total;// N*D (proj residual, then MLP out)
    float* dinv = tmp + (size_t)total; // N   (deg -> dinv in place)
    float* norm = dinv + N;            // E
    float* cs   = norm + E;            // D (colsum)
    // cs2 = cs + DD, contiguous

    const int gN  = (N + 255) / 256;
    const int gE  = (E + 255) / 256;
    const int gT  = (total + 255) / 256;
    const int gEw = (E + 7) / 8;          // 1 wave per edge
    const int gRow = (N + 15) / 16;       // gemm blocks
    const int gCol = (N + 63) / 64;       // colsum blocks
    const int gHead = (N + 7) / 8;

    // degrees -> dinv -> per-edge norm
    k_init_deg<<<gN, 256, 0, stream>>>(dinv, N);
    k_add_deg<<<gE, 256, 0, stream>>>(ecol, dinv, E);
    k_make_dinv<<<gN, 256, 0, stream>>>(dinv, N);
    k_make_norm<<<gE, 256, 0, stream>>>(erow, ecol, dinv, norm, E);

    // layer-0 residual: tmp = xin @ proj_w
    gemm_wmma_f32<<<gRow, 256, 0, stream>>>(xin, proj_w, nullptr, tmp, N, 0,
                                            nullptr, nullptr, nullptr);

    for (int i = 0; i < L; ++i) {
        const float* xi  = (i == 0) ? xin : xcur;
        const float* res = (i == 0) ? tmp : xcur;
        // hw = xi @ conv_w[i]; fused: agg = hw*dinv^2 + conv_b[i]
        gemm_wmma_f32<<<gRow, 256, 0, stream>>>(xi, conv_w + (size_t)i * DD * DD,
                                                nullptr, hw, N, 0,
                                                dinv, conv_b + (size_t)i * DD, agg);
        k_edge_scatter<<<gEw, 256, 0, stream>>>(erow, ecol, norm, hw, agg, E);
        // graph norm
        k_zero<<<1, 256, 0, stream>>>(cs, 2 * DD);
        k_colsum<<<gCol, 256, 0, stream>>>(agg, cs, N);
        k_center<<<gCol, 256, 0, stream>>>(agg, cs, gn_ms + (size_t)i * DD, cs + DD, N, inv_n);
        k_finalize<<<gT, 256, 0, stream>>>(agg, cs + DD, gn_w + (size_t)i * DD,
                                           gn_b + (size_t)i * DD, res, xcur, total, inv_n);
    }

    // tmp = relu(xcur @ mlp_w + mlp_b)
    gemm_wmma_f32<<<gRow, 256, 0, stream>>>(xcur, mlp_w, mlp_b, tmp, N, 1,
                                            nullptr, nullptr, nullptr);
    // out = tmp @ head_w + head_b
    k_head<<<gHead, 256, 0, stream>>>(tmp, head_w, head_b, out, N);
}